// renderModule_60473139528047
// MI455X (gfx1250) — compile-verified
//
#include <hip/hip_runtime.h>

typedef __attribute__((ext_vector_type(2))) float v2f;
typedef __attribute__((ext_vector_type(8))) float v8f;

namespace {
constexpr int   kH    = 1024;
constexpr int   kW    = 1024;
constexpr float kNear = 0.05f;
}

// ---------------------------------------------------------------------------
// Zero the accumulators (col_acc lives in d_out, w_acc in d_ws).
// ---------------------------------------------------------------------------
__global__ void zero_acc_kernel(float* __restrict__ col, int ncol,
                                float* __restrict__ wac, int nw) {
  int i      = blockIdx.x * blockDim.x + threadIdx.x;
  int stride = gridDim.x * blockDim.x;
  for (int k = i; k < ncol; k += stride) col[k] = 0.0f;
  for (int k = i; k < nw;   k += stride) wac[k] = 0.0f;
}

// ---------------------------------------------------------------------------
// Splat kernel. Each wave32 processes 16 points per iteration:
//   A (16x4)  = rows of [R | t]   (rows 0..2 nonzero)
//   B (4x16)  = 16 points as homogeneous columns (x,y,z,1)
//   D (16x16) = rows 0..2 are pc_x / pc_y / pc_z of point n at lane n.
// Lanes l and l+16 both own point (l&15): low half does the dv=0 taps,
// high half the dv=1 taps -> all 32 lanes issue atomics.
// ---------------------------------------------------------------------------
__global__ __launch_bounds__(256) void splat_kernel(
    const float* __restrict__ cam, const float* __restrict__ pts,
    float* __restrict__ colacc, float* __restrict__ wacc, int npts) {
  const int      lane   = threadIdx.x & 31;
  const int      m      = lane & 15;
  const bool     hi     = (lane >= 16);
  const unsigned wave   = (unsigned)(blockIdx.x * blockDim.x + threadIdx.x) >> 5;
  const unsigned nwaves = (unsigned)(gridDim.x * blockDim.x) >> 5;

  const float fx = cam[0] * (float)kW;
  const float fy = cam[1] * (float)kH;
  const float cx = cam[2] * (float)kW;
  const float cy = cam[3] * (float)kH;

  // A-matrix (16x4 f32): lane holds row M=lane%16; VGPR0 = K0 (lo) / K2 (hi),
  // VGPR1 = K1 (lo) / K3 (hi).  Row m<3: [R[m][0..2], t[m]], else zero.
  float a0 = 0.0f, a1 = 0.0f;
  if (m < 3) {
    a0 = hi ? cam[7 + 3 * m + 2] : cam[7 + 3 * m + 0];
    a1 = hi ? cam[4 + m]         : cam[7 + 3 * m + 1];
  }
  v2f A;
  A.x = a0;
  A.y = a1;

  const unsigned nbatch = ((unsigned)npts + 15u) >> 4;
  for (unsigned b = wave; b < nbatch; b += nwaves) {
    const unsigned p        = (b << 4) + (unsigned)m;
    const bool     in_range = (p < (unsigned)npts);
    const unsigned pidx     = in_range ? p : (unsigned)(npts - 1);
    const float* __restrict__ pt = pts + (size_t)pidx * 7u;

    // Prefetch the point this wave touches next iteration.
    const unsigned pn = p + (nwaves << 4);
    if (pn < (unsigned)npts) __builtin_prefetch(pts + (size_t)pn * 7u, 0, 0);

    // Unconditional loads from the clamped (always-valid) row; mask after.
    const float X  = pt[0], Y  = pt[1], Z  = pt[2];
    const float cr = pt[3], cg = pt[4], cb = pt[5];
    const float pw = in_range ? pt[6] : 0.0f;  // load stays unconditional:
    // clamped address is valid; only the value is selected.

    // B-matrix (4x16 f32): row K striped across lanes; lanes 0-15 carry
    // rows K=0 (VGPR0) / K=1 (VGPR1), lanes 16-31 rows K=2 / K=3.
    v2f B;
    B.x = hi ? Z : X;
    B.y = hi ? 1.0f : Y;

    v8f C = {};
    v8f D = __builtin_amdgcn_wmma_f32_16x16x4_f32(false, A, false, B,
                                                  (short)0, C, false, false);

    // D[m'][n]: VGPR m'%8, lane (m'/8)*16 + n.  Rows 0..2, n in 0..15.
    const float px = __shfl(D[0], m);
    const float py = __shfl(D[1], m);
    const float pz = __shfl(D[2], m);

    const float z     = pz + 1.0f;
    const bool  valid = in_range && (z > kNear);
    const float zs    = valid ? z : 1.0f;
    const float izs   = __builtin_amdgcn_rcpf(zs);  // one v_rcp, two muls
    const float u     = fx * px * izs + cx;
    const float v     = fy * py * izs + cy;
    const float u0    = floorf(u);
    const float v0    = floorf(v);
    const float fu    = u - u0;
    const float fv    = v - v0;
    const int   iu0   = (int)u0;
    const int   iv0   = (int)v0;

    const int   iv   = iv0 + (hi ? 1 : 0);
    const float fvt  = hi ? fv : (1.0f - fv);
    const bool  vrow = valid && (iv >= 0) && (iv < kH);

#pragma unroll
    for (int du = 0; du < 2; ++du) {
      const int   iu = iu0 + du;
      const float bw = (du ? fu : (1.0f - fu)) * fvt;
      if (vrow && iu >= 0 && iu < kW) {
        const float  wgt = pw * bw;
        const size_t q   = (size_t)iv * (size_t)kW + (size_t)iu;
        float* cp        = colacc + q * 3u;
        atomicAdd(cp + 0, cr * wgt);
        atomicAdd(cp + 1, cg * wgt);
        atomicAdd(cp + 2, cb * wgt);
        atomicAdd(wacc + q, wgt);
      }
    }
  }
}

// ---------------------------------------------------------------------------
// Composite: out = (col_acc + env) / (w_acc + 1).  In place in d_out.
// ---------------------------------------------------------------------------
__global__ void finalize_kernel(const float* __restrict__ env,
                                const float* __restrict__ wacc,
                                float* __restrict__ out, int npix) {
  int i      = blockIdx.x * blockDim.x + threadIdx.x;
  int stride = gridDim.x * blockDim.x;
  for (int p = i; p < npix; p += stride) {
    const float ws = wacc[p] + 1.0f;
    const float r  = (out[p * 3 + 0] + env[p * 3 + 0]) / ws;
    const float g  = (out[p * 3 + 1] + env[p * 3 + 1]) / ws;
    const float bl = (out[p * 3 + 2] + env[p * 3 + 2]) / ws;
    out[p * 3 + 0] = r;
    out[p * 3 + 1] = g;
    out[p * 3 + 2] = bl;
  }
}

// ---------------------------------------------------------------------------
// Inputs (setup_inputs order): [0] cam_type (int,1), [1] camera (f32,16),
// [2] points (f32, N*7), [3] environment (f32, H*W*3).
// d_out: H*W*3 f32.  d_ws: w_acc (H*W f32 = 4 MB).
// ---------------------------------------------------------------------------
extern "C" void kernel_launch(void* const* d_in, const int* in_sizes, int n_in,
                              void* d_out, int out_size, void* d_ws,
                              size_t ws_size, hipStream_t stream) {
  (void)n_in;
  (void)out_size;
  (void)ws_size;
  const float* cam = (const float*)d_in[1];
  const float* pts = (const float*)d_in[2];
  const float* env = (const float*)d_in[3];
  float*       out = (float*)d_out;
  float*       wac = (float*)d_ws;

  const int npts = in_sizes[2] / 7;
  const int npix = kH * kW;

  zero_acc_kernel<<<1024, 256, 0, stream>>>(out, npix * 3, wac, npix);
  splat_kernel<<<4096, 256, 0, stream>>>(cam, pts, out, wac, npts);
  finalize_kernel<<<2048, 256, 0, stream>>>(env, wac, out, npix);
}